// EmetMamba_27805618274650
// MI455X (gfx1250) — compile-verified
//
#include <hip/hip_runtime.h>
#include <math.h>

// ---------------- model dims (from reference) ----------------
#define B_   4
#define L_   2048
#define DM_  128
#define ED_  256          // EF*DM
#define N_   16
#define DR_  8
#define K_   4
#define NL_  2
#define BL_  (B_ * L_)    // 8192 rows
#define LDBC 48           // padded dbc row stride (40 real cols + 8 pad)

typedef __attribute__((ext_vector_type(2))) float v2f;
typedef __attribute__((ext_vector_type(8))) float v8f;

// =====================================================================
// Depthwise causal conv (K=4) + instance-norm over DM + leaky-relu + residual
// one block per (b,l) position, 128 threads = DM channels
// =====================================================================
__global__ void conv_in_block_kernel(const float* __restrict__ xin,
                                     const float* __restrict__ w,     // (DM,K)
                                     const float* __restrict__ bias,  // (DM)
                                     float* __restrict__ xout) {
    const int r = blockIdx.x;          // b*L + l
    const int l = r % L_;
    const int c = threadIdx.x;         // channel 0..127
    __shared__ float s1[DM_];
    __shared__ float s2[DM_];

    float acc = bias[c];
#pragma unroll
    for (int k = 0; k < K_; ++k) {
        int li = l - (K_ - 1) + k;
        if (li >= 0)
            acc += w[c * K_ + k] * xin[(size_t)(r - (K_ - 1) + k) * DM_ + c];
    }

    s1[c] = acc;
    s2[c] = acc * acc;
    __syncthreads();
    for (int s = DM_ / 2; s > 0; s >>= 1) {
        if (c < s) { s1[c] += s1[c + s]; s2[c] += s2[c + s]; }
        __syncthreads();
    }
    const float mu  = s1[0] * (1.0f / DM_);
    const float var = s2[0] * (1.0f / DM_) - mu * mu;

    float vn = (acc - mu) * rsqrtf(var + 1e-5f);
    float lr = vn > 0.0f ? vn : 0.01f * vn;          // leaky_relu(0.01)
    xout[(size_t)r * DM_ + c] = lr + xin[(size_t)r * DM_ + c];  // residual
}

// =====================================================================
// RMSNorm over DM: out = x * rsqrt(mean(x^2)+eps) * w
// =====================================================================
__global__ void rmsnorm_kernel(const float* __restrict__ x,
                               const float* __restrict__ w,   // (DM)
                               float* __restrict__ out) {
    const int r = blockIdx.x;
    const int c = threadIdx.x;
    __shared__ float s1[DM_];
    float v = x[(size_t)r * DM_ + c];
    s1[c] = v * v;
    __syncthreads();
    for (int s = DM_ / 2; s > 0; s >>= 1) {
        if (c < s) s1[c] += s1[c + s];
        __syncthreads();
    }
    float ms = s1[0] * (1.0f / DM_);
    out[(size_t)r * DM_ + c] = v * rsqrtf(ms + 1e-5f) * w[c];
}

// =====================================================================
// Zero-pad x_proj_w (ED x 40) into (ED x 48)
// =====================================================================
__global__ void pad_xproj_w_kernel(const float* __restrict__ w,
                                   float* __restrict__ wp) {
    const int idx = blockIdx.x * blockDim.x + threadIdx.x;   // ED*48
    if (idx >= ED_ * LDBC) return;
    const int row = idx / LDBC;
    const int col = idx % LDBC;
    wp[idx] = (col < DR_ + 2 * N_) ? w[row * (DR_ + 2 * N_) + col] : 0.0f;
}

// =====================================================================
// fp32 GEMM via V_WMMA_F32_16X16X4_F32 (wave32), software-pipelined.
// One wave computes a 32 x (NT*16) tile: 2 M-subtiles x NT N-subtiles,
// so each B fragment feeds 2 WMMAs and each A fragment feeds NT WMMAs.
// Double-buffered K-loop: fragments for k0+4 are loaded before the
// WMMA batch for k0, overlapping load latency with matrix math.
// NO bounds checks: caller guarantees M%32==0, N==Ngroups*NT*16.
//   A: M x K (row-major, lda)   fragment: M = lane%16, K pair = (lane/16)*2
//   B: K x N (row-major, ldb)   fragment: N = lane%16, same K pair
//   C: M x N (ldc)              D layout: row = vgpr + 8*(lane/16)
// accum==1 -> C += A*B
// =====================================================================
template <int NT>
__global__ void gemm_wmma_f32(const float* __restrict__ A,
                              const float* __restrict__ Bm,
                              float* __restrict__ C,
                              int Mtiles32, int Ngroups, int K,
                              int lda, int ldb, int ldc, int accum) {
    const int wave = (int)((blockIdx.x * blockDim.x + threadIdx.x) >> 5);
    const int lane = threadIdx.x & 31;
    if (wave >= Mtiles32 * Ngroups) return;

    const int mt = wave / Ngroups;
    const int ng = wave % Ngroups;
    const int mrow0 = (mt << 5) + (lane & 15);     // A row, M-subtile 0
    const int koff  = (lane >> 4) << 1;            // K offset {0,2} by lane half
    const int nb    = ng * (NT * 16) + (lane & 15);

    const float* Ap = A + (size_t)mrow0 * lda + koff;          // subtile 0
    const float* Aq = Ap + (size_t)16 * lda;                   // subtile 1
    const float* Bp = Bm + (size_t)koff * ldb + nb;

    v8f acc0[NT] = {};
    v8f acc1[NT] = {};

    // prologue: load fragments for k0 = 0
    v2f a0c, a1c, bc[NT];
    a0c.x = Ap[0]; a0c.y = Ap[1];
    a1c.x = Aq[0]; a1c.y = Aq[1];
#pragma unroll
    for (int j = 0; j < NT; ++j) { bc[j].x = Bp[j * 16]; bc[j].y = Bp[j * 16 + ldb]; }

    for (int k0 = 0; k0 + 4 < K; k0 += 4) {
        Ap += 4; Aq += 4; Bp += (size_t)4 * ldb;
        // issue next iteration's loads first (overlap with WMMA below)
        v2f a0n, a1n, bn[NT];
        a0n.x = Ap[0]; a0n.y = Ap[1];
        a1n.x = Aq[0]; a1n.y = Aq[1];
#pragma unroll
        for (int j = 0; j < NT; ++j) { bn[j].x = Bp[j * 16]; bn[j].y = Bp[j * 16 + ldb]; }
        // compute on current fragments
#pragma unroll
        for (int j = 0; j < NT; ++j)
            acc0[j] = __builtin_amdgcn_wmma_f32_16x16x4_f32(
                false, a0c, false, bc[j], (short)0, acc0[j], false, false);
#pragma unroll
        for (int j = 0; j < NT; ++j)
            acc1[j] = __builtin_amdgcn_wmma_f32_16x16x4_f32(
                false, a1c, false, bc[j], (short)0, acc1[j], false, false);
        a0c = a0n; a1c = a1n;
#pragma unroll
        for (int j = 0; j < NT; ++j) bc[j] = bn[j];
    }
    // epilogue: last K-step
#pragma unroll
    for (int j = 0; j < NT; ++j)
        acc0[j] = __builtin_amdgcn_wmma_f32_16x16x4_f32(
            false, a0c, false, bc[j], (short)0, acc0[j], false, false);
#pragma unroll
    for (int j = 0; j < NT; ++j)
        acc1[j] = __builtin_amdgcn_wmma_f32_16x16x4_f32(
            false, a1c, false, bc[j], (short)0, acc1[j], false, false);

    // store: D row = rbase + vgpr, col striped by lane%16
    const int rbase0 = (mt << 5) + ((lane >> 4) << 3);
    const int rbase1 = rbase0 + 16;
#pragma unroll
    for (int j = 0; j < NT; ++j) {
        const int col = ng * (NT * 16) + j * 16 + (lane & 15);
#pragma unroll
        for (int rr = 0; rr < 8; ++rr) {
            size_t i0 = (size_t)(rbase0 + rr) * ldc + col;
            size_t i1 = (size_t)(rbase1 + rr) * ldc + col;
            float v0 = acc0[j][rr];
            float v1 = acc1[j][rr];
            C[i0] = accum ? (C[i0] + v0) : v0;
            C[i1] = accum ? (C[i1] + v1) : v1;
        }
    }
}

// =====================================================================
// Depthwise causal conv (K=4) on u-half of xz (stride 2*ED) + SiLU -> u
// =====================================================================
__global__ void mconv_silu_kernel(const float* __restrict__ xz,   // (BL, 2*ED)
                                  const float* __restrict__ w,    // (ED,K)
                                  const float* __restrict__ bias, // (ED)
                                  float* __restrict__ u) {        // (BL, ED)
    const int idx = blockIdx.x * blockDim.x + threadIdx.x;
    if (idx >= BL_ * ED_) return;
    const int c = idx % ED_;
    const int r = idx / ED_;
    const int l = r % L_;
    float acc = bias[c];
#pragma unroll
    for (int k = 0; k < K_; ++k) {
        int li = l - (K_ - 1) + k;
        if (li >= 0)
            acc += w[c * K_ + k] * xz[(size_t)(r - (K_ - 1) + k) * (2 * ED_) + c];
    }
    float sg = 1.0f / (1.0f + expf(-acc));
    u[(size_t)r * ED_ + c] = acc * sg;     // silu
}

// =====================================================================
// delta = softplus(delta_pre + bias[ed])   (in place, BL x ED)
// =====================================================================
__global__ void softplus_bias_kernel(float* __restrict__ delta,
                                     const float* __restrict__ bias) {
    const int idx = blockIdx.x * blockDim.x + threadIdx.x;
    if (idx >= BL_ * ED_) return;
    const int c = idx % ED_;
    float v = delta[idx] + bias[c];
    delta[idx] = (v > 20.0f) ? v : log1pf(expf(v));
}

// =====================================================================
// Selective scan.  thread = (b, ed, n), lane%16 == n so the sum over N
// is 4 shfl_xor steps inside a 16-lane group.  Each thread runs the
// L=2048 recurrence:  h = exp(dt*A)*h + dt*u*B ; y = sum_n h*C.
// Fuses  y = (scan + u*D) * silu(z)  into the writeback.
// grid: B * (ED/16) blocks of 256 threads (16 ed x 16 n per block)
// =====================================================================
__global__ void scan_kernel(const float* __restrict__ u,      // (BL,ED)
                            const float* __restrict__ delta,  // (BL,ED)
                            const float* __restrict__ dbc,    // (BL,LDBC) cols 8..23=B, 24..39=C
                            const float* __restrict__ xz,     // (BL,2*ED), z at col ED+ed
                            const float* __restrict__ A_log,  // (ED,N)
                            const float* __restrict__ Dp,     // (ED)
                            float* __restrict__ y) {          // (BL,ED)
    const int b   = blockIdx.x >> 4;        // 0..3
    const int edg = blockIdx.x & 15;        // 0..15
    const int n   = threadIdx.x & 15;
    const int ed  = edg * 16 + (threadIdx.x >> 4);

    const float Aval = -expf(A_log[ed * N_ + n]);
    const float Dv   = Dp[ed];
    float h = 0.0f;
    const size_t rbase = (size_t)b * L_;

    for (int t = 0; t < L_; ++t) {
        const size_t r = rbase + t;
        const float dt = delta[r * ED_ + ed];
        const float ut = u[r * ED_ + ed];
        const float Bt = dbc[r * LDBC + DR_ + n];
        const float Ct = dbc[r * LDBC + DR_ + N_ + n];

        h = expf(dt * Aval) * h + dt * ut * Bt;

        float yc = h * Ct;
        yc += __shfl_xor(yc, 8, 16);
        yc += __shfl_xor(yc, 4, 16);
        yc += __shfl_xor(yc, 2, 16);
        yc += __shfl_xor(yc, 1, 16);

        if (n == 0) {
            const float z  = xz[r * (2 * ED_) + ED_ + ed];
            const float sg = z / (1.0f + expf(-z));          // silu(z)
            y[r * ED_ + ed] = (yc + ut * Dv) * sg;
        }
    }
}

// =====================================================================
// Host side
// =====================================================================
template <int NT>
static inline void launch_gemm(const float* A, const float* Bm, float* C,
                               int M, int N, int K, int lda, int ldb, int ldc,
                               int accum, hipStream_t stream) {
    int Mtiles32 = M / 32;
    int Ngroups  = N / (NT * 16);           // exact by construction
    int waves    = Mtiles32 * Ngroups;
    int blocks   = (waves + 7) / 8;         // 8 waves (256 thr) per block
    gemm_wmma_f32<NT><<<blocks, 256, 0, stream>>>(A, Bm, C, Mtiles32, Ngroups, K,
                                                  lda, ldb, ldc, accum);
}

extern "C" void kernel_launch(void* const* d_in, const int* in_sizes, int n_in,
                              void* d_out, int out_size, void* d_ws, size_t ws_size,
                              hipStream_t stream) {
    (void)in_sizes; (void)n_in; (void)out_size; (void)ws_size;

    const float* x          = (const float*)d_in[0];
    const float* conv_w     = (const float*)d_in[1];   // (S,DM,K)
    const float* conv_b     = (const float*)d_in[2];   // (S,DM)
    const float* in_proj_w  = (const float*)d_in[3];   // (NL,DM,2*ED)
    const float* mconv_w    = (const float*)d_in[4];   // (NL,ED,K)
    const float* mconv_b    = (const float*)d_in[5];   // (NL,ED)
    const float* x_proj_w   = (const float*)d_in[6];   // (NL,ED,40)
    const float* dt_proj_w  = (const float*)d_in[7];   // (NL,DR,ED)
    const float* dt_proj_b  = (const float*)d_in[8];   // (NL,ED)
    const float* A_log      = (const float*)d_in[9];   // (NL,ED,N)
    const float* D_param    = (const float*)d_in[10];  // (NL,ED)
    const float* out_proj_w = (const float*)d_in[11];  // (NL,ED,DM)
    const float* norm_w     = (const float*)d_in[12];  // (NL,DM)

    float* xout = (float*)d_out;                       // x lives here (B,L,DM)
    float* ws   = (float*)d_ws;

    // workspace layout (floats)
    float* bufA = ws;                 // 1,048,576  (BL*DM)
    float* bufB = ws + 1048576;       // 1,048,576  (BL*DM)  conv pong / hin
    float* xz   = ws + 2097152;       // 4,194,304  (BL*2*ED)
    float* ubuf = ws + 6291456;       // 2,097,152  (BL*ED)
    float* dbuf = ws + 8388608;       // 2,097,152  (BL*ED)  delta
    float* ybuf = ws + 10485760;      // 2,097,152  (BL*ED)
    float* dbc  = ws + 12582912;      //   393,216  (BL*48, padded)
    float* wpad = ws + 12976128;      //    12,288  (ED*48, padded x_proj_w)

    // ---- 3 conv/IN/residual blocks (sequential) ----
    conv_in_block_kernel<<<BL_, DM_, 0, stream>>>(x,    conv_w + 0 * DM_ * K_, conv_b + 0 * DM_, bufA);
    conv_in_block_kernel<<<BL_, DM_, 0, stream>>>(bufA, conv_w + 1 * DM_ * K_, conv_b + 1 * DM_, bufB);
    conv_in_block_kernel<<<BL_, DM_, 0, stream>>>(bufB, conv_w + 2 * DM_ * K_, conv_b + 2 * DM_, xout);

    const int ne = BL_ * ED_;
    for (int l = 0; l < NL_; ++l) {
        // hin = rmsnorm(x) * w
        rmsnorm_kernel<<<BL_, DM_, 0, stream>>>(xout, norm_w + l * DM_, bufB);
        // xz = hin @ in_proj_w[l]     (8192 x 128 -> 512), N=512 = 8 groups * 64
        launch_gemm<4>(bufB, in_proj_w + (size_t)l * DM_ * 2 * ED_, xz,
                       BL_, 2 * ED_, DM_, DM_, 2 * ED_, 2 * ED_, 0, stream);
        // u = silu(dwconv(xz[:, :ED]))
        mconv_silu_kernel<<<(ne + 255) / 256, 256, 0, stream>>>(
            xz, mconv_w + l * ED_ * K_, mconv_b + l * ED_, ubuf);
        // pad x_proj_w[l] (256x40) -> wpad (256x48, zeros in cols 40..47)
        pad_xproj_w_kernel<<<(ED_ * LDBC + 255) / 256, 256, 0, stream>>>(
            x_proj_w + (size_t)l * ED_ * (DR_ + 2 * N_), wpad);
        // dbc = u @ wpad              (8192 x 256 -> 48), N=48 = 1 group * 48
        launch_gemm<3>(ubuf, wpad, dbc,
                       BL_, LDBC, ED_, ED_, LDBC, LDBC, 0, stream);
        // delta_pre = dbc[:, :8] @ dt_proj_w[l]   (8192 x 8 -> 256)
        launch_gemm<4>(dbc, dt_proj_w + (size_t)l * DR_ * ED_, dbuf,
                       BL_, ED_, DR_, LDBC, ED_, ED_, 0, stream);
        // delta = softplus(delta_pre + b)
        softplus_bias_kernel<<<(ne + 255) / 256, 256, 0, stream>>>(dbuf, dt_proj_b + l * ED_);
        // y = selective_scan(...) * silu(z)
        scan_kernel<<<B_ * (ED_ / 16), 256, 0, stream>>>(
            ubuf, dbuf, dbc, xz, A_log + l * ED_ * N_, D_param + l * ED_, ybuf);
        // x += y @ out_proj_w[l]      (8192 x 256 -> 128, accumulate)
        launch_gemm<4>(ybuf, out_proj_w + (size_t)l * ED_ * DM_, xout,
                       BL_, DM_, ED_, ED_, DM_, DM_, 1, stream);
    }
}